// NegativeLogLikelihoodLoss_43868795961969
// MI455X (gfx1250) — compile-verified
//
#include <hip/hip_runtime.h>
#include <hip/hip_bf16.h>

// NLL of diagonal-block multivariate Gaussian, B=256,S=512,J=32,D=3.
// Memory-bound streaming reduction: 252MB in, 1 float out (~10.8us @ 23.3TB/s).
// Strategy: TDM (tensor_load_to_lds) double-buffered streaming into LDS,
// compute from LDS, deterministic two-pass reduction (no float atomics).

#define ITEMS_PER_TILE 512
#define TILES_PER_BLOCK 8
#define ITEMS_PER_BLOCK (ITEMS_PER_TILE * TILES_PER_BLOCK)  // 4096
#define NBLOCKS 1024
#define NTHREADS 256
#define TOTAL_ITEMS (NBLOCKS * ITEMS_PER_BLOCK)  // 4,194,304 == 256*512*32

// Per-tile LDS layout (floats): [d:1536][mu:1536][L:4608] = 7680 floats = 30KB
#define TILE_FLOATS 7680
#define D_OFF  0
#define MU_OFF 1536
#define L_OFF  3072

typedef unsigned int v4u __attribute__((ext_vector_type(4)));
typedef int          v8i __attribute__((ext_vector_type(8)));
typedef int          v4i __attribute__((ext_vector_type(4)));

// Issue one TDM 1-D contiguous copy: n dwords from global gsrc -> LDS byte
// offset lds_off. Descriptor per CDNA5 ISA ch.8:
//   group0: count=1, lds_addr, global_addr[56:0], type=2
//   group1: data_size=4B; tensor = n x 1; tile = n x 1
__device__ __forceinline__ void tdm_load_1d(const void* gsrc, unsigned lds_off,
                                            unsigned n /*dwords*/) {
  unsigned long long ga = (unsigned long long)gsrc;
  v4u g0;
  g0.x = 1u;                                                   // count=1 (valid)
  g0.y = lds_off;                                              // lds_addr (bytes)
  g0.z = (unsigned)(ga & 0xFFFFFFFFull);                       // global_addr lo
  g0.w = ((unsigned)(ga >> 32) & 0x01FFFFFFu) | 0x80000000u;   // addr hi | type=2

  v8i g1;
  g1.s0 = (int)0x00020000u;                               // data_size=2 (4B), mask=0
  g1.s1 = (int)((n & 0xFFFFu) << 16);                     // tensor_dim0[15:0]
  g1.s2 = (int)(((n >> 16) & 0xFFFFu) | (1u << 16));      // dim0[31:16] | dim1[15:0]=1
  g1.s3 = (int)((n & 0xFFFFu) << 16);                     // dim1[31:16]=0 | tile_dim0
  g1.s4 = 1;                                              // tile_dim1=1, tile_dim2=0
  g1.s5 = (int)n;                                         // dim0_stride[31:0]
  g1.s6 = (int)(((n >> 16) & 0xFFFFu) | ((n & 0xFFFFu) << 16)); // d0s[47:32]|d1s[15:0]
  g1.s7 = (int)(n >> 16);                                 // dim1_stride[47:16]

  v4i z4 = {0, 0, 0, 0};
  v8i z8 = {0, 0, 0, 0, 0, 0, 0, 0};
  // amdgpu-toolchain (clang-23) 6-arg form: (g0, g1, g2, g3, extra, cpol)
  __builtin_amdgcn_tensor_load_to_lds(g0, g1, z4, z4, z8, 0);
}

__global__ __launch_bounds__(NTHREADS)
void nll_partial_kernel(const float* __restrict__ y_true,
                        const float* __restrict__ y_mean,
                        const float* __restrict__ Lmat,
                        float* __restrict__ partials) {
  __shared__ float smem[2 * TILE_FLOATS];   // 60KB double buffer
  __shared__ float sred[NTHREADS / 32];

  const int tid = threadIdx.x;
  const long long blockItem0 = (long long)blockIdx.x * ITEMS_PER_BLOCK;
  const unsigned smemBase = (unsigned)(unsigned long long)(void*)smem;

  const bool issuer = (tid < 32);  // wave 0 drives the TDM

  // Prologue: tile 0 -> buffer 0 (3 DMAs)
  if (issuer) {
    const long long it0 = blockItem0;
    tdm_load_1d(y_true + it0 * 3, smemBase + D_OFF * 4,  ITEMS_PER_TILE * 3);
    tdm_load_1d(y_mean + it0 * 3, smemBase + MU_OFF * 4, ITEMS_PER_TILE * 3);
    tdm_load_1d(Lmat   + it0 * 9, smemBase + L_OFF * 4,  ITEMS_PER_TILE * 9);
  }

  float acc = 0.0f;
  for (int t = 0; t < TILES_PER_BLOCK; ++t) {
    if (issuer) {
      if (t + 1 < TILES_PER_BLOCK) {
        const long long it = blockItem0 + (long long)(t + 1) * ITEMS_PER_TILE;
        const unsigned boff = smemBase + (unsigned)(((t + 1) & 1) * TILE_FLOATS * 4);
        tdm_load_1d(y_true + it * 3, boff + D_OFF * 4,  ITEMS_PER_TILE * 3);
        tdm_load_1d(y_mean + it * 3, boff + MU_OFF * 4, ITEMS_PER_TILE * 3);
        tdm_load_1d(Lmat   + it * 9, boff + L_OFF * 4,  ITEMS_PER_TILE * 9);
        // 3 ops of tile t+1 outstanding; wait for tile t's 3 to retire (in-order)
        __builtin_amdgcn_s_wait_tensorcnt(3);
      } else {
        __builtin_amdgcn_s_wait_tensorcnt(0);
      }
    }
    __syncthreads();  // publish tile t's LDS data to all waves

    const float* buf = smem + (t & 1) * TILE_FLOATS;
    const float* dT  = buf + D_OFF;
    const float* mu  = buf + MU_OFF;
    const float* Lb  = buf + L_OFF;

#pragma unroll
    for (int k = 0; k < ITEMS_PER_TILE / NTHREADS; ++k) {
      const int i = tid + k * NTHREADS;
      // strides 3 and 9 dwords are odd -> bank-conflict-free across 32 lanes
      const float d0 = dT[i * 3 + 0] - mu[i * 3 + 0];
      const float d1 = dT[i * 3 + 1] - mu[i * 3 + 1];
      const float d2 = dT[i * 3 + 2] - mu[i * 3 + 2];
      const float* l = Lb + i * 9;
      const float L00 = l[0], L10 = l[3], L11 = l[4];
      const float L20 = l[6], L21 = l[7], L22 = l[8];
      // forward substitution L x = d
      const float x0 = d0 / L00;
      const float x1 = (d1 - L10 * x0) / L11;
      const float x2 = (d2 - L20 * x0 - L21 * x1) / L22;
      const float mah = x0 * x0 + x1 * x1 + x2 * x2;
      const float logdet = 2.0f * __logf(L00 * L11 * L22);
      acc += 0.5f * (mah + logdet);
    }
    __syncthreads();  // buffer (t&1) free to be overwritten next iteration
  }

  // wave32 reduction
  for (int off = 16; off > 0; off >>= 1) acc += __shfl_down(acc, off, 32);
  if ((tid & 31) == 0) sred[tid >> 5] = acc;
  __syncthreads();
  if (tid == 0) {
    float s = 0.0f;
#pragma unroll
    for (int w = 0; w < NTHREADS / 32; ++w) s += sred[w];
    partials[blockIdx.x] = s;
  }
}

__global__ __launch_bounds__(NTHREADS)
void nll_final_kernel(const float* __restrict__ partials, float* __restrict__ out) {
  __shared__ float sred[NTHREADS / 32];
  const int tid = threadIdx.x;
  float s = 0.0f;
  for (int i = tid; i < NBLOCKS; i += NTHREADS) s += partials[i];
  for (int off = 16; off > 0; off >>= 1) s += __shfl_down(s, off, 32);
  if ((tid & 31) == 0) sred[tid >> 5] = s;
  __syncthreads();
  if (tid == 0) {
    float tot = 0.0f;
#pragma unroll
    for (int w = 0; w < NTHREADS / 32; ++w) tot += sred[w];
    const float LOG_2PI = 1.8378770664093453f;
    out[0] = tot / (float)TOTAL_ITEMS + 0.5f * 3.0f * LOG_2PI;
  }
}

extern "C" void kernel_launch(void* const* d_in, const int* in_sizes, int n_in,
                              void* d_out, int out_size, void* d_ws, size_t ws_size,
                              hipStream_t stream) {
  const float* y_true = (const float*)d_in[0];
  const float* y_mean = (const float*)d_in[1];
  const float* Lmat   = (const float*)d_in[2];
  float* partials = (float*)d_ws;  // NBLOCKS floats = 4KB scratch

  nll_partial_kernel<<<NBLOCKS, NTHREADS, 0, stream>>>(y_true, y_mean, Lmat, partials);
  nll_final_kernel<<<1, NTHREADS, 0, stream>>>(partials, (float*)d_out);
}